// N_Loss_65266323030640
// MI455X (gfx1250) — compile-verified
//
#include <hip/hip_runtime.h>

typedef float v2f __attribute__((ext_vector_type(2)));
typedef float v8f __attribute__((ext_vector_type(8)));

constexpr int N   = 8192;
constexpr int D   = 64;
constexpr int BLK = 64;                       // 64x64 output block per wave
constexpr int NB  = N / BLK;                  // 128 block rows
constexpr int NPAIR_BLOCKS = NB * (NB + 1) / 2;   // 8256 upper-tri block pairs
constexpr int WAVES_PER_WG = 8;                   // 256 threads, wave32
constexpr int NUM_WG = NPAIR_BLOCKS / WAVES_PER_WG;  // 1032 exactly

__global__ void zero_out_kernel(float* out) { out[0] = 0.0f; }

__global__ __launch_bounds__(256)
void pair_bce_kernel(const float* __restrict__ X,
                     const int*   __restrict__ target,
                     float*       __restrict__ out) {
    const int lane    = threadIdx.x & 31;
    const int wave    = threadIdx.x >> 5;
    const int w       = blockIdx.x * WAVES_PER_WG + wave;
    const int halfsel = lane >> 4;   // 0 for lanes 0-15, 1 for lanes 16-31
    const int l15     = lane & 15;

    // Map linear wave id -> (bi, bj) with bi <= bj over NB x NB blocks.
    int bi = 0, rem = w;
    while (rem >= NB - bi) { rem -= NB - bi; ++bi; }
    const int bj    = bi + rem;
    const int ibase = bi * BLK;
    const int jbase = bj * BLK;

    // f32 WMMA 16x16x4 operand layout: lane L supplies row (base + (L&15));
    // per K-step s the two consecutive columns k0 = s*4 + (L>>4)*2, k0+1
    // (VGPR0 = k0, VGPR1 = k0+1) -> one aligned float2 load per fragment.
    const float* Arow[4];
    const float* Brow[4];
    #pragma unroll
    for (int t = 0; t < 4; ++t) {
        Arow[t] = X + (size_t)(ibase + t * 16 + l15) * D;
        Brow[t] = X + (size_t)(jbase + t * 16 + l15) * D;
    }

    v8f acc[4][4];
    const v8f vzero = {};
    #pragma unroll
    for (int a = 0; a < 4; ++a)
        #pragma unroll
        for (int b = 0; b < 4; ++b)
            acc[a][b] = vzero;

    // K loop: D=64 in 16 steps of K=4; 16 WMMAs per step (4x4 register tile).
    #pragma unroll
    for (int s = 0; s < 16; ++s) {
        const int k0 = s * 4 + halfsel * 2;
        v2f afrag[4], bfrag[4];
        #pragma unroll
        for (int t = 0; t < 4; ++t) {
            afrag[t] = *(const v2f*)(Arow[t] + k0);
            bfrag[t] = *(const v2f*)(Brow[t] + k0);
        }
        #pragma unroll
        for (int tm = 0; tm < 4; ++tm)
            #pragma unroll
            for (int tn = 0; tn < 4; ++tn)
                acc[tm][tn] = __builtin_amdgcn_wmma_f32_16x16x4_f32(
                    /*neg_a=*/false, afrag[tm],
                    /*neg_b=*/false, bfrag[tn],
                    /*c_mod=*/(short)0, acc[tm][tn],
                    /*reuse_a=*/false, /*reuse_b=*/false);
    }

    // Loss phase. C/D layout: VGPR r holds row M = r + 8*(lane>>4), col = lane&15.
    // Per-pair loss: softplus(x) - y*x, stable softplus via
    //   max(x,0) + log(1 + exp(-|x|))   using native v_exp_f32 / v_log_f32.
    float lsum = 0.0f;
    #pragma unroll
    for (int tm = 0; tm < 4; ++tm) {
        // Row targets for this 16-row tile (independent of tn) - load once.
        int ti[8];
        #pragma unroll
        for (int r = 0; r < 8; ++r)
            ti[r] = target[ibase + tm * 16 + r + halfsel * 8];

        #pragma unroll
        for (int tn = 0; tn < 4; ++tn) {
            const int gj = jbase + tn * 16 + l15;
            const int tj = target[gj];
            #pragma unroll
            for (int r = 0; r < 8; ++r) {
                const int gi = ibase + tm * 16 + r + halfsel * 8;
                if (gi < gj) {
                    const float x  = acc[tm][tn][r];
                    const float sp = fmaxf(x, 0.0f)
                                   + __logf(1.0f + __expf(-fabsf(x)));
                    lsum += (ti[r] == tj) ? (sp - x) : sp;
                }
            }
        }
    }

    // Wave32 butterfly reduction.
    #pragma unroll
    for (int off = 16; off > 0; off >>= 1)
        lsum += __shfl_xor(lsum, off, 32);

    __shared__ float wsum[WAVES_PER_WG];
    if (lane == 0) wsum[wave] = lsum;
    __syncthreads();

    if (threadIdx.x == 0) {
        float s = 0.0f;
        #pragma unroll
        for (int i = 0; i < WAVES_PER_WG; ++i) s += wsum[i];
        constexpr float inv_npairs = (float)(2.0 / ((double)N * (double)(N - 1)));
#if defined(__HIP_DEVICE_COMPILE__)
        unsafeAtomicAdd(out, s * inv_npairs);   // global_atomic_add_f32
#else
        atomicAdd(out, s * inv_npairs);
#endif
    }
}

extern "C" void kernel_launch(void* const* d_in, const int* in_sizes, int n_in,
                              void* d_out, int out_size, void* d_ws, size_t ws_size,
                              hipStream_t stream) {
    const float* X   = (const float*)d_in[0];   // [8192, 64] fp32
    const int*   tgt = (const int*)d_in[1];     // [8192] int
    float*       out = (float*)d_out;           // scalar fp32

    zero_out_kernel<<<1, 1, 0, stream>>>(out);
    pair_bce_kernel<<<NUM_WG, 256, 0, stream>>>(X, tgt, out);
}